// GAT_82901458747986
// MI455X (gfx1250) — compile-verified
//
#include <hip/hip_runtime.h>
#include <math.h>

#define NN 50000       // nodes
#define NE 200000      // raw edges
#define ET 250000      // edges + self loops
#define IN_CH 14
#define KP 16          // padded K for WMMA
#define HC 1024        // HEADS*HID
#define HH 8
#define FD 128
#define OD 4
#define NEG_SLOPE 0.2f

typedef __attribute__((ext_vector_type(2))) float v2f;
typedef __attribute__((ext_vector_type(8))) float v8f;

// ---------------- helpers ----------------

__device__ __forceinline__ void edge_sd(const long long* __restrict__ ei, int e, int& s, int& d) {
    if (e < NE) { s = (int)ei[e]; d = (int)ei[NE + e]; }
    else        { s = e - NE; d = e - NE; }            // appended self-loops
}

// float atomic-max via sign-split int punning (valid with -inf init)
__device__ __forceinline__ void atomicMaxFloat(float* addr, float val) {
    if (val >= 0.f) atomicMax((int*)addr, __float_as_int(val));
    else            atomicMin((unsigned int*)addr, __float_as_uint(val));
}

// ---------------- pad x -> [NN,16], W1 -> [16,HC] (zeros in K=14,15) ----------------
__global__ void k_pad(const float* __restrict__ x, const float* __restrict__ W1,
                      float* __restrict__ xp, float* __restrict__ w1p) {
    int i = blockIdx.x * blockDim.x + threadIdx.x;
    if (i < NN * KP) {
        int r = i >> 4, k = i & 15;
        xp[i] = (k < IN_CH) ? x[r * IN_CH + k] : 0.f;
    }
    if (i < KP * HC) w1p[i] = (i < IN_CH * HC) ? W1[i] : 0.f;
}

// ---------------- layer-1 GEMM: h = xpad @ W1pad (WMMA f32 16x16x4) ----------------
// 3125 M-tiles x 64 N-tiles = 200000 tiles, one wave32 per tile, 8 waves/block.
// No guards, no divergence: 50000%16==0, 1024%16==0, K padded to 16.
__global__ void k_gemm1(const float* __restrict__ xp, const float* __restrict__ w1p,
                        float* __restrict__ h) {
    int wid  = blockIdx.x * 8 + (threadIdx.x >> 5);   // global wave id, uniform per wave
    int lane = threadIdx.x & 31;
    int mt = wid >> 6;           // 0..3124
    int nt = wid & 63;           // 0..63
    int half = lane >> 4;        // 0: lanes 0-15 (K pair k0,k0+1), 1: lanes 16-31 (k0+2,k0+3)
    int l    = lane & 15;
    int row = mt * 16 + l;       // A: M index for this lane
    int col = nt * 16 + l;       // B: N index for this lane

    const v2f* ap = (const v2f*)(xp + (size_t)row * KP);   // 8B-aligned pairs
    v8f c = {};
#pragma unroll
    for (int kk = 0; kk < 4; ++kk) {
        int k0 = kk * 4 + half * 2;                   // this lane's K pair (even)
        v2f a = ap[k0 >> 1];
        v2f b;
        b.x = w1p[(size_t)k0       * HC + col];
        b.y = w1p[(size_t)(k0 + 1) * HC + col];
        c = __builtin_amdgcn_wmma_f32_16x16x4_f32(false, a, false, b,
                                                  (short)0, c, false, false);
    }
    // C/D layout: VGPR v -> row M0 + v + 8*half, col N0 + l
    float* hp = h + (size_t)(mt * 16 + half * 8) * HC + nt * 16 + l;
#pragma unroll
    for (int v = 0; v < 8; ++v) hp[(size_t)v * HC] = c[v];
}

// ---------------- per-node attention coefficients, layer 1 ----------------
__global__ void k_alpha1(const float* __restrict__ h, const float* __restrict__ att_s,
                         const float* __restrict__ att_d,
                         float* __restrict__ as1, float* __restrict__ ad1) {
    int n = blockIdx.x, t = threadIdx.x;               // 128 threads
    __shared__ float rs[128], rd[128];
    const float* hn = h + (size_t)n * HC;
    for (int head = 0; head < HH; ++head) {
        float v = hn[head * FD + t];
        rs[t] = v * att_s[head * FD + t];
        rd[t] = v * att_d[head * FD + t];
        __syncthreads();
        for (int s = 64; s > 0; s >>= 1) {
            if (t < s) { rs[t] += rs[t + s]; rd[t] += rd[t + s]; }
            __syncthreads();
        }
        if (t == 0) { as1[n * HH + head] = rs[0]; ad1[n * HH + head] = rd[0]; }
        __syncthreads();
    }
}

// ---------------- init kernels ----------------
__global__ void k_zero(float* __restrict__ p, long long n) {
    long long i = (long long)blockIdx.x * blockDim.x + threadIdx.x;
    if (i < n) p[i] = 0.f;
}

__global__ void k_init_small(float* __restrict__ m1, float* __restrict__ s1,
                             float* __restrict__ m2, float* __restrict__ s2,
                             float* __restrict__ out, const float* __restrict__ b2) {
    int i = blockIdx.x * blockDim.x + threadIdx.x;
    if (i < NN * HH) { m1[i] = -INFINITY; s1[i] = 0.f; }
    if (i < NN)      { m2[i] = -INFINITY; s2[i] = 0.f; }
    if (i < NN * OD) { out[i] = b2[i & 3]; }           // seed output with bias b2
}

// ---------------- edge phase, layer 1 ----------------
__global__ void k_logits1(const long long* __restrict__ ei, const float* __restrict__ as1,
                          const float* __restrict__ ad1, float* __restrict__ logits,
                          float* __restrict__ m1) {
    int idx = blockIdx.x * blockDim.x + threadIdx.x;   // ET*HH
    if (idx >= ET * HH) return;
    int e = idx >> 3, head = idx & 7;
    int s, d; edge_sd(ei, e, s, d);
    float z = as1[s * HH + head] + ad1[d * HH + head];
    z = z > 0.f ? z : NEG_SLOPE * z;
    logits[idx] = z;
    atomicMaxFloat(&m1[d * HH + head], z);
}

__global__ void k_exps1(const long long* __restrict__ ei, float* __restrict__ logits,
                        const float* __restrict__ m1, float* __restrict__ s1) {
    int idx = blockIdx.x * blockDim.x + threadIdx.x;
    if (idx >= ET * HH) return;
    int e = idx >> 3, head = idx & 7;
    int s, d; edge_sd(ei, e, s, d);
    float ex = expf(logits[idx] - m1[d * HH + head]);
    logits[idx] = ex;
    atomicAdd(&s1[d * HH + head], ex);
}

__global__ void k_scatter1(const long long* __restrict__ ei, const float* __restrict__ h,
                           const float* __restrict__ logits, const float* __restrict__ s1,
                           float* __restrict__ out1) {
    int e = blockIdx.x, t = threadIdx.x;               // 256 threads, one block per edge
    int s, d; edge_sd(ei, e, s, d);
    __shared__ float w[HH];
    if (t < HH) w[t] = logits[e * HH + t] / (s1[d * HH + t] + 1e-16f);
    __syncthreads();
    const float* hs = h    + (size_t)s * HC;
    float*       od = out1 + (size_t)d * HC;
#pragma unroll
    for (int i = 0; i < 4; ++i) {
        int f = i * 256 + t;
        atomicAdd(&od[f], hs[f] * w[f >> 7]);
    }
}

__global__ void k_bias_elu(float* __restrict__ out1, const float* __restrict__ b1) {
    long long i = (long long)blockIdx.x * blockDim.x + threadIdx.x;
    if (i >= (long long)NN * HC) return;
    float v = out1[i] + b1[i & (HC - 1)];
    out1[i] = v > 0.f ? v : expm1f(v);
}

// ---------------- layer-2 linear + attention coefficients ----------------
__global__ void k_lin2(const float* __restrict__ h1, const float* __restrict__ W2,
                       const float* __restrict__ a_s2, const float* __restrict__ a_d2,
                       float* __restrict__ h2, float* __restrict__ as2,
                       float* __restrict__ ad2) {
    int n = blockIdx.x, t = threadIdx.x;               // 128 threads
    float p[OD] = {0.f, 0.f, 0.f, 0.f};
    const float* hn = h1 + (size_t)n * HC;
    for (int k = t; k < HC; k += 128) {
        float hk = hn[k];
#pragma unroll
        for (int j = 0; j < OD; ++j) p[j] += hk * W2[k * OD + j];
    }
    __shared__ float red[OD][128];
#pragma unroll
    for (int j = 0; j < OD; ++j) red[j][t] = p[j];
    __syncthreads();
    for (int s = 64; s > 0; s >>= 1) {
        if (t < s) {
#pragma unroll
            for (int j = 0; j < OD; ++j) red[j][t] += red[j][t + s];
        }
        __syncthreads();
    }
    if (t == 0) {
        float ss = 0.f, dd = 0.f;
#pragma unroll
        for (int j = 0; j < OD; ++j) {
            float v = red[j][0];
            h2[n * OD + j] = v;
            ss += v * a_s2[j];
            dd += v * a_d2[j];
        }
        as2[n] = ss; ad2[n] = dd;
    }
}

// ---------------- edge phase, layer 2 (1 head) ----------------
__global__ void k_logits2(const long long* __restrict__ ei, const float* __restrict__ as2,
                          const float* __restrict__ ad2, float* __restrict__ logits,
                          float* __restrict__ m2) {
    int e = blockIdx.x * blockDim.x + threadIdx.x;
    if (e >= ET) return;
    int s, d; edge_sd(ei, e, s, d);
    float z = as2[s] + ad2[d];
    z = z > 0.f ? z : NEG_SLOPE * z;
    logits[e] = z;
    atomicMaxFloat(&m2[d], z);
}

__global__ void k_exps2(const long long* __restrict__ ei, float* __restrict__ logits,
                        const float* __restrict__ m2, float* __restrict__ s2) {
    int e = blockIdx.x * blockDim.x + threadIdx.x;
    if (e >= ET) return;
    int s, d; edge_sd(ei, e, s, d);
    float ex = expf(logits[e] - m2[d]);
    logits[e] = ex;
    atomicAdd(&s2[d], ex);
}

__global__ void k_scatter2(const long long* __restrict__ ei, const float* __restrict__ h2,
                           const float* __restrict__ logits, const float* __restrict__ s2,
                           float* __restrict__ out) {
    int e = blockIdx.x * blockDim.x + threadIdx.x;
    if (e >= ET) return;
    int s, d; edge_sd(ei, e, s, d);
    float w = logits[e] / (s2[d] + 1e-16f);
#pragma unroll
    for (int j = 0; j < OD; ++j)
        atomicAdd(&out[d * OD + j], h2[s * OD + j] * w);
}

// ---------------- launch ----------------
extern "C" void kernel_launch(void* const* d_in, const int* in_sizes, int n_in,
                              void* d_out, int out_size, void* d_ws, size_t ws_size,
                              hipStream_t stream) {
    const float*     x    = (const float*)d_in[0];
    const long long* ei   = (const long long*)d_in[1];   // int64 [2, NE]
    const float*     W1   = (const float*)d_in[2];
    const float*     aS1  = (const float*)d_in[3];
    const float*     aD1  = (const float*)d_in[4];
    const float*     b1   = (const float*)d_in[5];
    const float*     W2   = (const float*)d_in[6];
    const float*     aS2  = (const float*)d_in[7];
    const float*     aD2  = (const float*)d_in[8];
    const float*     b2   = (const float*)d_in[9];
    float* out = (float*)d_out;

    // workspace partition (floats)
    float* ws      = (float*)d_ws;
    float* h       = ws;                          // 51,200,000
    float* out1    = ws + 51200000LL;             // 51,200,000
    float* as1     = ws + 102400000LL;            // 400,000
    float* ad1     = ws + 102800000LL;            // 400,000
    float* m1      = ws + 103200000LL;            // 400,000
    float* s1      = ws + 103600000LL;            // 400,000
    float* logits1 = ws + 104000000LL;            // 2,000,000
    float* h2      = ws + 106000000LL;            // 200,000
    float* as2     = ws + 106200000LL;            // 50,000
    float* ad2     = ws + 106250000LL;            // 50,000
    float* m2      = ws + 106300000LL;            // 50,000
    float* s2      = ws + 106350000LL;            // 50,000
    float* logits2 = ws + 106400000LL;            // 250,000
    float* xpad    = ws + 106650000LL;            // 800,000   [NN,16]
    float* w1pad   = ws + 107450000LL;            // 16,384    [16,HC]

    // pad inputs for branch-free WMMA GEMM
    k_pad<<<(NN * KP + 255) / 256, 256, 0, stream>>>(x, W1, xpad, w1pad);
    // layer 1 GEMM: 3125*64 tiles / 8 waves per block
    k_gemm1<<<25000, 256, 0, stream>>>(xpad, w1pad, h);
    // per-node attention dots
    k_alpha1<<<NN, 128, 0, stream>>>(h, aS1, aD1, as1, ad1);
    // init accumulators
    k_zero<<<(int)(((long long)NN * HC + 255) / 256), 256, 0, stream>>>(out1, (long long)NN * HC);
    k_init_small<<<(NN * HH + 255) / 256, 256, 0, stream>>>(m1, s1, m2, s2, out, b2);
    // edge phase 1
    k_logits1<<<(ET * HH + 255) / 256, 256, 0, stream>>>(ei, as1, ad1, logits1, m1);
    k_exps1<<<(ET * HH + 255) / 256, 256, 0, stream>>>(ei, logits1, m1, s1);
    k_scatter1<<<ET, 256, 0, stream>>>(ei, h, logits1, s1, out1);
    // bias + ELU
    k_bias_elu<<<(int)(((long long)NN * HC + 255) / 256), 256, 0, stream>>>(out1, b1);
    // layer 2
    k_lin2<<<NN, 128, 0, stream>>>(out1, W2, aS2, aD2, h2, as2, ad2);
    k_logits2<<<(ET + 255) / 256, 256, 0, stream>>>(ei, as2, ad2, logits2, m2);
    k_exps2<<<(ET + 255) / 256, 256, 0, stream>>>(ei, logits2, m2, s2);
    k_scatter2<<<(ET + 255) / 256, 256, 0, stream>>>(ei, h2, logits2, s2, out);
}